// RightPool_68710886801779
// MI455X (gfx1250) — compile-verified
//
#include <hip/hip_runtime.h>

// RightPool (cummax along width) for x: (16, 256, 128, 128) f32.
// Memory-bound: 512 MiB traffic -> ~23 us floor at 23.3 TB/s.
// One wave32 per 128-float row; 4 elems/lane; async global->LDS staging
// (gfx1250 GLOBAL_LOAD_ASYNC_TO_LDS_B128 + s_wait_asynccnt) and a
// ds_bpermute-based wave32 max-scan.

#define WIDTH            128
#define WAVES_PER_BLOCK  8
#define BLOCK_THREADS    256   // 8 wave32s

__global__ __launch_bounds__(BLOCK_THREADS)
void rightpool_cummax_kernel(const float* __restrict__ x,
                             float* __restrict__ out,
                             int nrows) {
    // One 128-float row per wave: 4 KB total, disjoint 16B slot per thread.
    __shared__ float tile[BLOCK_THREADS * 4];

    const int tid  = threadIdx.x;
    const int lane = tid & 31;
    const int wave = tid >> 5;
    const int row  = blockIdx.x * WAVES_PER_BLOCK + wave;
    if (row >= nrows) return;   // wave-uniform; grid divides exactly anyway

    const size_t g  = (size_t)row * WIDTH + (size_t)(lane << 2);
    const float* gp = x + g;

    // Low 32 bits of a flat pointer into LDS == group-relative LDS byte
    // offset (ISA 10.2 aperture mapping), which is exactly what the async
    // instruction's VDST operand wants (hw adds LDS_BASE).
    const unsigned lds_off = (unsigned)(size_t)(&tile[tid << 2]);

    // Async DMA 16B/lane from global into this thread's LDS slot.
    asm volatile("global_load_async_to_lds_b128 %0, %1, off"
                 :
                 : "v"(lds_off), "v"(gp)
                 : "memory");
    // Wave-private counter; each wave reads only LDS it wrote -> no barrier.
    asm volatile("s_wait_asynccnt 0" ::: "memory");

    const float4 a = *(const float4*)(&tile[tid << 2]);

    // Lane-local inclusive prefix max over 4 contiguous elements.
    const float p0 = a.x;
    const float p1 = fmaxf(p0, a.y);
    const float p2 = fmaxf(p1, a.z);
    const float p3 = fmaxf(p2, a.w);

    // Wave32 inclusive max-scan of per-lane totals (Hillis-Steele, 5 steps;
    // __shfl_up lowers to ds_bpermute_b32 on gfx1250 wave32).
    float t = p3;
    #pragma unroll
    for (int d = 1; d < 32; d <<= 1) {
        const float u = __shfl_up(t, d, 32);
        if (lane >= d) t = fmaxf(t, u);
    }
    // Exclusive prefix for this lane = inclusive scan of lane-1.
    float e = __shfl_up(t, 1, 32);
    if (lane == 0) e = -__builtin_inff();

    float4 o;
    o.x = fmaxf(p0, e);
    o.y = fmaxf(p1, e);
    o.z = fmaxf(p2, e);
    o.w = fmaxf(p3, e);

    *(float4*)(out + g) = o;
}

extern "C" void kernel_launch(void* const* d_in, const int* in_sizes, int n_in,
                              void* d_out, int out_size, void* d_ws, size_t ws_size,
                              hipStream_t stream) {
    (void)n_in; (void)out_size; (void)d_ws; (void)ws_size;
    const float* x   = (const float*)d_in[0];
    float*       out = (float*)d_out;

    const int total = in_sizes[0];            // 16*256*128*128 = 67,108,864
    const int nrows = total / WIDTH;          // 524,288 rows
    const int blocks = (nrows + WAVES_PER_BLOCK - 1) / WAVES_PER_BLOCK;  // 65,536

    rightpool_cummax_kernel<<<blocks, BLOCK_THREADS, 0, stream>>>(x, out, nrows);
}